// DualPathMamba_83425444758243
// MI455X (gfx1250) — compile-verified
//
#include <hip/hip_runtime.h>

// CDNA5 (gfx1250) fused dual-path Mamba.
// f32 WMMA (V_WMMA_F32_16X16X4_F32) for all projections, wave32 shuffles for
// the scan reduction, one workgroup per sequence, chunked LDS pipeline.

typedef __attribute__((ext_vector_type(2))) float v2f;
typedef __attribute__((ext_vector_type(8))) float v8f;

#define D_MODEL 64
#define D_INNER 128
#define D_STATE 16
#define DT_RANK 4
#define D_CONV  4
#define CH      64
#define NTHREADS 256

// padded LDS strides (bank-decorrelated for WMMA-A reads, keep 8B alignment)
#define UNS 68
#define XCS 132
#define YSS 132
#define OBS 68
#define DBS 48

static constexpr int SMEM_FLOATS =
    CH*D_MODEL       // u_s
  + CH*UNS           // un_s
  + CH*OBS           // ob_s
  + CH*2*D_INNER     // xz_s
  + CH*XCS           // xc_s
  + CH*DBS           // dbl_s
  + CH*D_INNER       // dtb_s
  + CH*YSS           // y_s
  + D_INNER*D_STATE  // A_s
  + D_INNER          // Dp_s
  + D_INNER          // dt_b_s
  + D_INNER*DT_RANK  // dt_w_s
  + D_MODEL          // gamma_s
  + D_INNER*D_CONV   // conv_w_s
  + D_INNER          // conv_b_s
  + 3*D_INNER;       // conv carry

__device__ __forceinline__ float silu_f(float v) {
  return v * (1.0f / (1.0f + __expf(-v)));
}

template<int PASS>
__global__ __launch_bounds__(NTHREADS, 1)
void mamba_pass_kernel(const float* __restrict__ xin, float* __restrict__ xout,
                       const float* __restrict__ gamma_g, const float* __restrict__ in_w_g,
                       const float* __restrict__ conv_w_g, const float* __restrict__ conv_b_g,
                       const float* __restrict__ xproj_w_g, const float* __restrict__ dt_w_g,
                       const float* __restrict__ dt_b_g, const float* __restrict__ A_log_g,
                       const float* __restrict__ Dp_g, const float* __restrict__ out_w_g)
{
  extern __shared__ float smem[];
  constexpr int Bn = 4, F = 256, T = 384;
  constexpr int L = (PASS == 0) ? F : T;
  constexpr int NCHUNK = L / CH;

  float* u_s   = smem;
  float* un_s  = u_s   + CH*D_MODEL;
  float* ob_s  = un_s  + CH*UNS;
  float* xz_s  = ob_s  + CH*OBS;
  float* xc_s  = xz_s  + CH*2*D_INNER;
  float* dbl_s = xc_s  + CH*XCS;
  float* dtb_s = dbl_s + CH*DBS;
  float* y_s   = dtb_s + CH*D_INNER;
  float* A_s   = y_s   + CH*YSS;
  float* Dp_s  = A_s   + D_INNER*D_STATE;
  float* dtbia = Dp_s  + D_INNER;
  float* dtw_s = dtbia + D_INNER;
  float* gam_s = dtw_s + D_INNER*DT_RANK;
  float* cw_s  = gam_s + D_MODEL;
  float* cb_s  = cw_s  + D_INNER*D_CONV;
  float* carry = cb_s  + D_INNER;

  const int tid  = threadIdx.x;
  const int lane = tid & 31;
  const int wave = tid >> 5;
  const int row  = lane & 15;   // A row / B col / D col within tile
  const int half = lane >> 4;   // K sub-pair select & D row-half

  const int s = blockIdx.x;
  const int bIdx     = s / ((PASS == 0) ? T : F);
  const int fixedIdx = s % ((PASS == 0) ? T : F);  // PASS0: t position; PASS1: f position

  const float* inw = in_w_g    + PASS * (2*D_INNER*D_MODEL);
  const float* xpw = xproj_w_g + PASS * ((DT_RANK + 2*D_STATE)*D_INNER);
  const float* oww = out_w_g   + PASS * (D_MODEL*D_INNER);

  // ---- preload small params into LDS ----
  for (int e = tid; e < D_INNER*D_STATE; e += NTHREADS)
    A_s[e] = -__expf(A_log_g[PASS*D_INNER*D_STATE + e]);
  for (int e = tid; e < D_INNER; e += NTHREADS) {
    Dp_s[e]  = Dp_g[PASS*D_INNER + e];
    dtbia[e] = dt_b_g[PASS*D_INNER + e];
    cb_s[e]  = conv_b_g[PASS*D_INNER + e];
  }
  for (int e = tid; e < D_INNER*DT_RANK; e += NTHREADS) {
    dtw_s[e] = dt_w_g[PASS*D_INNER*DT_RANK + e];
    cw_s[e]  = conv_w_g[PASS*D_INNER*D_CONV + e];
  }
  for (int e = tid; e < D_MODEL; e += NTHREADS)
    gam_s[e] = gamma_g[PASS*D_MODEL + e];
  for (int e = tid; e < 3*D_INNER; e += NTHREADS)
    carry[e] = 0.0f;

  // persistent scan state: thread owns (d = tid>>1, 8 of 16 n-states)
  float h[8];
  #pragma unroll
  for (int j = 0; j < 8; ++j) h[j] = 0.0f;
  const int sd = tid >> 1;
  const int nh = (tid & 1) * 8;

  __syncthreads();

  for (int ck = 0; ck < NCHUNK; ++ck) {
    const int tok0 = ck * CH;

    // ---- load residual/input chunk (L2-absorbed gather) ----
    for (int e = tid; e < CH*D_MODEL; e += NTHREADS) {
      int tc = e >> 6, c = e & 63;
      int tg = tok0 + tc;
      long addr;
      if constexpr (PASS == 0)
        addr = (((long)(bIdx*D_MODEL + c) * F + tg) * T + fixedIdx);   // x(B,C,F,T)
      else
        addr = (((long)(bIdx*T + tg) * F + fixedIdx) * D_MODEL + c);   // ws(B,T,F,C)
      u_s[tc*D_MODEL + c] = xin[addr];
    }
    __syncthreads();

    // ---- RMSNorm (4 lanes per token, shuffle-combine) ----
    {
      int tc = tid >> 2, q = (tid & 3) * 16;
      float ss = 0.f;
      #pragma unroll
      for (int k = 0; k < 16; ++k) { float v = u_s[tc*D_MODEL + q + k]; ss += v*v; }
      ss += __shfl_xor(ss, 1, 32);
      ss += __shfl_xor(ss, 2, 32);
      float sc = 8.0f / fmaxf(sqrtf(ss), 1e-12f);
      #pragma unroll
      for (int k = 0; k < 16; ++k) {
        int c = q + k;
        un_s[tc*UNS + c] = u_s[tc*D_MODEL + c] * sc * gam_s[c];
      }
    }
    __syncthreads();

    // ---- GEMM1: xz = un @ in_w.T   (64x64 x 64x256), f32 WMMA ----
    for (int tile = wave; tile < 4*16; tile += 8) {
      int mt = tile >> 4, nt = tile & 15;
      int m = mt*16 + row;
      int n = nt*16 + row;
      v8f acc = {};
      #pragma unroll
      for (int kb = 0; kb < D_MODEL; kb += 4) {
        int k = kb + half*2;
        v2f a  = *(const v2f*)&un_s[m*UNS + k];
        v2f bb = *(const v2f*)&inw[n*D_MODEL + k];
        acc = __builtin_amdgcn_wmma_f32_16x16x4_f32(false, a, false, bb,
                                                    (short)0, acc, false, false);
      }
      #pragma unroll
      for (int r = 0; r < 8; ++r)
        xz_s[(mt*16 + r + half*8)*(2*D_INNER) + nt*16 + row] = acc[r];
    }
    __syncthreads();

    // ---- depthwise causal conv4 + bias + SiLU ----
    for (int e = tid; e < CH*D_INNER; e += NTHREADS) {
      int tc = e >> 7, d = e & 127;
      float acc = cb_s[d];
      #pragma unroll
      for (int k = 0; k < D_CONV; ++k) {
        int idx = tc - 3 + k;
        float xv = (idx >= 0) ? xz_s[idx*(2*D_INNER) + d]
                              : carry[(idx + 3)*D_INNER + d];  // zero at seq start
        acc += cw_s[d*D_CONV + k] * xv;
      }
      xc_s[tc*XCS + d] = silu_f(acc);
    }
    __syncthreads();
    // save conv carry (raw x of last 3 tokens) for next chunk
    for (int e = tid; e < 3*D_INNER; e += NTHREADS) {
      int r = e >> 7, d = e & 127;
      carry[r*D_INNER + d] = xz_s[(CH - 3 + r)*(2*D_INNER) + d];
    }

    // ---- GEMM2: dbl = xc @ xproj_w.T   (64x128 x 128x36 -> padded 48) ----
    for (int tile = wave; tile < 4*3; tile += 8) {
      int mt = tile / 3, nt = tile % 3;
      int m = mt*16 + row;
      int n = nt*16 + row;
      bool nv = n < (DT_RANK + 2*D_STATE);
      v8f acc = {};
      #pragma unroll
      for (int kb = 0; kb < D_INNER; kb += 4) {
        int k = kb + half*2;
        v2f a  = *(const v2f*)&xc_s[m*XCS + k];
        v2f bb = {0.f, 0.f};
        if (nv) bb = *(const v2f*)&xpw[n*D_INNER + k];
        acc = __builtin_amdgcn_wmma_f32_16x16x4_f32(false, a, false, bb,
                                                    (short)0, acc, false, false);
      }
      #pragma unroll
      for (int r = 0; r < 8; ++r)
        dbl_s[(mt*16 + r + half*8)*DBS + nt*16 + row] = acc[r];
    }
    __syncthreads();

    // ---- dt = softplus(dbl[:, :4] @ dt_w.T + dt_b)  (K=4 -> scalar) ----
    for (int e = tid; e < CH*D_INNER; e += NTHREADS) {
      int tc = e >> 7, d = e & 127;
      float v = dtbia[d];
      #pragma unroll
      for (int r = 0; r < DT_RANK; ++r)
        v += dbl_s[tc*DBS + r] * dtw_s[d*DT_RANK + r];
      dtb_s[tc*D_INNER + d] = (v > 20.f) ? v : log1pf(__expf(v));
    }
    __syncthreads();

    // ---- selective scan: serial over tokens, 2048-wide over (d, n) ----
    #pragma unroll 1
    for (int tc = 0; tc < CH; ++tc) {
      float dtv = dtb_s[tc*D_INNER + sd];
      float xv  = xc_s[tc*XCS + sd];
      float dtx = dtv * xv;
      const float* Bp = &dbl_s[tc*DBS + DT_RANK];
      const float* Cp = &dbl_s[tc*DBS + DT_RANK + D_STATE];
      float acc = 0.f;
      #pragma unroll
      for (int j = 0; j < 8; ++j) {
        int n = nh + j;
        float dA = __expf(dtv * A_s[sd*D_STATE + n]);
        h[j] = dA * h[j] + dtx * Bp[n];
        acc += h[j] * Cp[n];
      }
      acc += __shfl_xor(acc, 1, 32);         // combine the two n-halves
      if ((tid & 1) == 0)
        y_s[tc*YSS + sd] = acc + xv * Dp_s[sd];
    }
    __syncthreads();

    // ---- gate: y *= silu(z) ----
    for (int e = tid; e < CH*D_INNER; e += NTHREADS) {
      int tc = e >> 7, d = e & 127;
      y_s[tc*YSS + d] *= silu_f(xz_s[tc*(2*D_INNER) + D_INNER + d]);
    }
    __syncthreads();

    // ---- GEMM3: out = u + y @ out_w.T   (64x128 x 128x64) ----
    for (int tile = wave; tile < 4*4; tile += 8) {
      int mt = tile >> 2, nt = tile & 3;
      int m = mt*16 + row;
      int n = nt*16 + row;
      v8f acc = {};
      #pragma unroll
      for (int kb = 0; kb < D_INNER; kb += 4) {
        int k = kb + half*2;
        v2f a  = *(const v2f*)&y_s[m*YSS + k];
        v2f bb = *(const v2f*)&oww[n*D_INNER + k];
        acc = __builtin_amdgcn_wmma_f32_16x16x4_f32(false, a, false, bb,
                                                    (short)0, acc, false, false);
      }
      #pragma unroll
      for (int r = 0; r < 8; ++r) {
        int mr = mt*16 + r + half*8;
        int cc = nt*16 + row;
        ob_s[mr*OBS + cc] = acc[r] + u_s[mr*D_MODEL + cc];
      }
    }
    __syncthreads();

    // ---- store chunk (layout per pass) ----
    for (int e = tid; e < CH*D_MODEL; e += NTHREADS) {
      int tc, c;
      if constexpr (PASS == 0) { tc = e >> 6; c = e & 63; }   // contiguous (f,c)
      else                     { tc = e & 63; c = e >> 6; }   // contiguous t per c
      int tg = tok0 + tc;
      long addr;
      if constexpr (PASS == 0)
        addr = (((long)(bIdx*T + fixedIdx) * F + tg) * D_MODEL + c);  // ws(B,T,F,C)
      else
        addr = (((long)(bIdx*D_MODEL + c) * F + fixedIdx) * T + tg);  // out(B,C,F,T)
      xout[addr] = ob_s[tc*OBS + c];
    }
    __syncthreads();
  }
}

extern "C" void kernel_launch(void* const* d_in, const int* in_sizes, int n_in,
                              void* d_out, int out_size, void* d_ws, size_t ws_size,
                              hipStream_t stream) {
  (void)in_sizes; (void)n_in; (void)out_size; (void)ws_size;
  const float* x      = (const float*)d_in[0];
  const float* gamma  = (const float*)d_in[1];
  const float* in_w   = (const float*)d_in[2];
  const float* conv_w = (const float*)d_in[3];
  const float* conv_b = (const float*)d_in[4];
  const float* xproj  = (const float*)d_in[5];
  const float* dt_w   = (const float*)d_in[6];
  const float* dt_b   = (const float*)d_in[7];
  const float* A_log  = (const float*)d_in[8];
  const float* Dp     = (const float*)d_in[9];
  const float* out_w  = (const float*)d_in[10];
  float* ws  = (float*)d_ws;    // staging (B,T,F,C) = 100.7 MB
  float* out = (float*)d_out;

  const size_t smem = (size_t)SMEM_FLOATS * sizeof(float);

  // pass 1: 4*384 = 1536 sequences of L=256 (over F)
  mamba_pass_kernel<0><<<4*384, NTHREADS, smem, stream>>>(
      x, ws, gamma, in_w, conv_w, conv_b, xproj, dt_w, dt_b, A_log, Dp, out_w);
  // pass 2: 4*256 = 1024 sequences of L=384 (over T)
  mamba_pass_kernel<1><<<4*256, NTHREADS, smem, stream>>>(
      ws, out, gamma, in_w, conv_w, conv_b, xproj, dt_w, dt_b, A_log, Dp, out_w);
}